// TempleGate_17008070492810
// MI455X (gfx1250) — compile-verified
//
#include <hip/hip_runtime.h>
#include <cstdint>
#include <cmath>

// ---------------------------------------------------------------------------
// TempleGate — MI455X (gfx1250)
// ~210MB HBM traffic -> ~9us roofline at 23.3 TB/s. One thread per source row
// (b,s): row async-copied to LDS (coalesced global_load_async_to_lds_b128),
// both extended rows + both gates emitted with 128-bit non-temporal stores.
// The 10-step recurrence collapses to dot(w, vec5) with w = breath^T @ temple
// (precomputed on-device into d_ws; branch is data-dependent via chaos).
// fmodf replaced by an exact constant-divisor floor-mod (~7 VALU vs ~75) so
// VALU stays under the memory roofline.
// ---------------------------------------------------------------------------

typedef float v4f __attribute__((ext_vector_type(4)));

#define B_DIM   256
#define T_DIM   4096
#define T2_DIM  8192
#define D_DIM   16
#define NSTEP   10
#define ROWS_PB 256
#define LDS_STRIDE_F 20   // 80B per row: 16B-aligned for b128, pads banks

#if defined(__AMDGCN__) && __has_builtin(__builtin_amdgcn_global_load_async_to_lds_b128)
#define HAVE_ASYNC_LDS 1
typedef int v4i __attribute__((ext_vector_type(4)));
typedef __attribute__((address_space(1))) v4i* as1_v4i;   // global int4*
typedef __attribute__((address_space(3))) v4i* as3_v4i;   // LDS int4*
#else
#define HAVE_ASYNC_LDS 0
#endif

// ws layout: [0..4]=w_main, [5..9]=w_mirror, [10]=branch (0=copy,1=retro,2=inv)
__global__ void tg_setup(const float* __restrict__ x,
                         const float* __restrict__ temple,
                         const float* __restrict__ bmain,
                         const float* __restrict__ bmir,
                         float* __restrict__ ws) {
  if (threadIdx.x == 0 && blockIdx.x == 0) {
    for (int j = 0; j < 5; ++j) {
      float wm = 0.f, wr = 0.f;
      for (int i = 0; i < 4; ++i) {
        wm += bmain[i] * temple[i * 5 + j];
        wr += bmir[i]  * temple[i * 5 + j];
      }
      ws[j]     = wm;
      ws[5 + j] = wr;
    }
    float x0    = x[0];
    float seed  = x0 - floorf(x0);                 // jnp.mod(x0, 1.0)
    float chaos = 3.5699456f * seed * (1.0f - seed);
    int branch  = (chaos > 0.7f) ? 2 : ((chaos > 0.4f) ? 1 : 0);
    ws[10] = (float)branch;
  }
}

// Exact floor-mod by 10 for |r| << 2^23: q = floor(r/10) approximated via
// reciprocal (off by at most 1 near boundaries), remainder via exact FMA,
// then two fixups select the true floor-mod in [0,10).
__device__ __forceinline__ float floormod10(float r) {
  float q = floorf(r * 0.1f);
  float m = fmaf(q, -10.0f, r);        // exact: both on common grid, |m|<20
  m = (m <  0.0f)  ? (m + 10.0f) : m;  // q one too large
  m = (m >= 10.0f) ? (m - 10.0f) : m;  // q one too small
  return m;
}

__device__ __forceinline__ float gate10(const float w[5],
                                        float a0, float a1, float a2,
                                        float a3, float a4) {
  float v0 = a0, v1 = a1, v2 = a2, v3 = a3, v4 = a4;
  float h[NSTEP];
#pragma unroll
  for (int k = 0; k < NSTEP; ++k) {
    float r = w[0] * v0 + w[1] * v1 + w[2] * v2 + w[3] * v3 + w[4] * v4;
    h[k] = r;                          // history stores raw resonance
    v4 = v3; v3 = v2; v2 = v1; v1 = v0;
    v0 = floormod10(r);                // roll + overwrite lane 0
  }
  float mean = 0.f;
#pragma unroll
  for (int k = 0; k < NSTEP; ++k) mean += h[k];
  mean *= (1.0f / NSTEP);
  float var = 0.f;
#pragma unroll
  for (int k = 0; k < NSTEP; ++k) { float d = h[k] - mean; var += d * d; }
  var *= (1.0f / (NSTEP - 1));         // ddof = 1
  return 1.0f / (1.0f + sqrtf(var));
}

__global__ __launch_bounds__(ROWS_PB) void tg_main(const float* __restrict__ x,
                                                   const float* __restrict__ ws,
                                                   float* __restrict__ out) {
  const int  tid = threadIdx.x;
  const long row = (long)blockIdx.x * ROWS_PB + tid;   // source row, exact fit

  float w_main[5], w_mir[5];
#pragma unroll
  for (int j = 0; j < 5; ++j) { w_main[j] = ws[j]; w_mir[j] = ws[5 + j]; }
  const int branch = (int)ws[10];

  v4f v[4];

#if HAVE_ASYNC_LDS
  // Stage the 256-row x tile into LDS with CDNA5 async copies: consecutive
  // lanes fetch consecutive 16B chunks -> fully coalesced HBM reads.
  __shared__ float tile[ROWS_PB * LDS_STRIDE_F];
  {
    const float* gsrc = x + (long)blockIdx.x * ROWS_PB * D_DIM;
#pragma unroll
    for (int it = 0; it < 4; ++it) {
      int c = tid + ROWS_PB * it;          // chunk 0..1023
      int r = c >> 2, k = c & 3;
      const float* s = gsrc + (long)c * 4;
      float*       d = &tile[r * LDS_STRIDE_F + k * 4];
      __builtin_amdgcn_global_load_async_to_lds_b128(
          (as1_v4i)(v4i*)s, (as3_v4i)(v4i*)d, 0, 0);
    }
  }
#if __has_builtin(__builtin_amdgcn_s_wait_asynccnt)
  __builtin_amdgcn_s_wait_asynccnt(0);
#else
  asm volatile("s_wait_asynccnt 0" ::: "memory");
#endif
  __syncthreads();
  {
    const v4f* lrow = (const v4f*)&tile[tid * LDS_STRIDE_F];
#pragma unroll
    for (int k = 0; k < 4; ++k) v[k] = lrow[k];
  }
#else
  {
    const v4f* grow = (const v4f*)(x + row * D_DIM);
#pragma unroll
    for (int k = 0; k < 4; ++k) v[k] = __builtin_nontemporal_load(&grow[k]);
  }
#endif

  const int b = (int)(row >> 12);            // T = 4096
  const int s = (int)(row & (T_DIM - 1));
  float* gate = out;
  float* ext  = out + (size_t)B_DIM * T2_DIM;

  // ---- main half: extended row t = s is a plain copy of x[b,s] ----
  {
    size_t er  = ((size_t)b << 13) | (size_t)s;
    v4f*  dst  = (v4f*)(ext + er * D_DIM);
#pragma unroll
    for (int k = 0; k < 4; ++k) __builtin_nontemporal_store(v[k], &dst[k]);
    float g = gate10(w_main, v[0].x, v[0].y, v[0].z, v[0].w, v[1].x);
    __builtin_nontemporal_store(g, &gate[er]);
  }

  // ---- mirror half: branch-dependent values and destination row ----
  {
    v4f m[4];
    if (branch == 2) {
#pragma unroll
      for (int k = 0; k < 4; ++k) m[k] = 9.0f - v[k];   // inverted
    } else {
#pragma unroll
      for (int k = 0; k < 4; ++k) m[k] = v[k];          // copy / retro values
    }
    // retro: x[b,s] lands at extended row 2T-1-s; else at T+s
    int dm    = (branch == 1) ? (T2_DIM - 1 - s) : (T_DIM + s);
    size_t er = ((size_t)b << 13) | (size_t)dm;
    v4f* dst  = (v4f*)(ext + er * D_DIM);
#pragma unroll
    for (int k = 0; k < 4; ++k) __builtin_nontemporal_store(m[k], &dst[k]);
    float g = gate10(w_mir, m[0].x, m[0].y, m[0].z, m[0].w, m[1].x);
    __builtin_nontemporal_store(g, &gate[er]);
  }
}

extern "C" void kernel_launch(void* const* d_in, const int* in_sizes, int n_in,
                              void* d_out, int out_size, void* d_ws, size_t ws_size,
                              hipStream_t stream) {
  const float* x      = (const float*)d_in[0];
  const float* temple = (const float*)d_in[1];
  const float* bmain  = (const float*)d_in[2];
  const float* bmir   = (const float*)d_in[3];
  float*       out    = (float*)d_out;
  float*       ws     = (float*)d_ws;

  tg_setup<<<1, 32, 0, stream>>>(x, temple, bmain, bmir, ws);

  const int blocks = (B_DIM * T_DIM) / ROWS_PB;   // 4096
  tg_main<<<blocks, ROWS_PB, 0, stream>>>(x, ws, out);
}